// MIUpperBound_14422500180352
// MI455X (gfx1250) — compile-verified
//
#include <hip/hip_runtime.h>

// MI455X / gfx1250, wave32.
// CLUB MI upper bound collapses algebraically:
//   result = S2/N - ||colsum||^2 / N^2
// Single pass over the 2048x64 fp32 input. Column reductions are driven
// through V_WMMA_F32_16X16X4_F32 with an all-ones A matrix: D = 1*B + C
// makes every row of D the running column-sum of the streamed 4x16 B tiles.
// A second WMMA on the element-squared tile accumulates sum-of-squares.

typedef __attribute__((ext_vector_type(2))) float v2f;
typedef __attribute__((ext_vector_type(8))) float v8f;

#define N_ROWS 2048
#define N_COLS 64
#define WAVES_PER_BLOCK 8
#define ROWS_PER_WAVE 32
#define NUM_WAVES (N_ROWS / ROWS_PER_WAVE)          // 64
#define GRID_X (NUM_WAVES / WAVES_PER_BLOCK)        // 8
#define WS_STRIDE 65                                // 64 colsums + 1 sumsq per wave

__global__ __launch_bounds__(256) void club_partial_kernel(
    const float* __restrict__ y, float* __restrict__ ws) {
  const int lane = threadIdx.x & 31;
  const int wave = threadIdx.x >> 5;
  const int gw   = blockIdx.x * WAVES_PER_BLOCK + wave;   // global wave id
  const int row0 = gw * ROWS_PER_WAVE;

  const int n    = lane & 15;   // column within a 16-wide group (B matrix N index)
  const int half = lane >> 4;   // lanes 16..31 carry the other two K rows

  v8f accS[4] = {};   // running column sums, one accumulator per 16-col group
  v8f accQ[4] = {};   // running column sums of squares

  v2f ones;
  ones.x = 1.0f;
  ones.y = 1.0f;

  for (int r = row0; r < row0 + ROWS_PER_WAVE; r += 4) {
    // B tile is 4 rows x 16 cols. Column n lives in lanes {n, n+16}, two
    // K elements per lane. With an all-ones A, any K placement sums correctly.
    const float* p0 = y + (size_t)(r + 2 * half) * N_COLS;
    const float* p1 = p0 + N_COLS;
#pragma unroll
    for (int g = 0; g < 4; ++g) {
      v2f b;
      b.x = p0[g * 16 + n];
      b.y = p1[g * 16 + n];
      accS[g] = __builtin_amdgcn_wmma_f32_16x16x4_f32(
          false, ones, false, b, (short)0, accS[g], false, false);
      v2f b2;
      b2.x = b.x * b.x;
      b2.y = b.y * b.y;
      accQ[g] = __builtin_amdgcn_wmma_f32_16x16x4_f32(
          false, ones, false, b2, (short)0, accQ[g], false, false);
    }
  }

  // All 16 rows of D are identical (= column sums). Element 0 of each lane:
  // lanes 0..15 hold D[0][lane], lanes 16..31 hold D[8][lane-16] (same value).
  float q = accQ[0][0] + accQ[1][0] + accQ[2][0] + accQ[3][0];
#pragma unroll
  for (int off = 16; off > 0; off >>= 1) q += __shfl_xor(q, off);
  // 32-lane sum double-counts every column once -> scale by 0.5.
  if (lane == 0) ws[gw * WS_STRIDE + 64] = 0.5f * q;
  if (lane < 16) {
#pragma unroll
    for (int g = 0; g < 4; ++g)
      ws[gw * WS_STRIDE + g * 16 + lane] = accS[g][0];
  }
}

__global__ void club_final_kernel(const float* __restrict__ ws,
                                  float* __restrict__ out) {
  __shared__ float sm[N_COLS];
  const int d = threadIdx.x;  // 64 threads, one per column
  float m = 0.0f;
  for (int w = 0; w < NUM_WAVES; ++w) m += ws[w * WS_STRIDE + d];
  sm[d] = m * m;
  __syncthreads();
  if (d == 0) {
    float m2 = 0.0f;
    for (int i = 0; i < N_COLS; ++i) m2 += sm[i];
    float s2 = 0.0f;
    for (int w = 0; w < NUM_WAVES; ++w) s2 += ws[w * WS_STRIDE + 64];
    const float invN = 1.0f / (float)N_ROWS;
    out[0] = s2 * invN - m2 * invN * invN;
  }
}

extern "C" void kernel_launch(void* const* d_in, const int* in_sizes, int n_in,
                              void* d_out, int out_size, void* d_ws, size_t ws_size,
                              hipStream_t stream) {
  (void)in_sizes; (void)n_in; (void)out_size; (void)ws_size;
  const float* y = (const float*)d_in[0];
  float* ws = (float*)d_ws;   // NUM_WAVES * 65 floats = 16640 bytes
  float* out = (float*)d_out;

  club_partial_kernel<<<GRID_X, 256, 0, stream>>>(y, ws);
  club_final_kernel<<<1, N_COLS, 0, stream>>>(ws, out);
}